// chamfer_loss_58437325029446
// MI455X (gfx1250) — compile-verified
//
#include <hip/hip_runtime.h>

typedef __attribute__((ext_vector_type(2))) float v2f;
typedef __attribute__((ext_vector_type(8))) float v8f;

#define BATCH      4
#define NBINS      256
#define MPTS       (240 * 320)      // 76800 targets per batch
#define MTILES     (MPTS / 16)      // 4800 column tiles per batch
#define WPB        150              // waves per batch
#define CTPW       (MTILES / WPB)   // 32 column tiles per wave
#define BIGF       1e10f
#define MIN_DEPTH  0.1f

// Direct-ISA min/max: the backend inserts v_max(v,v,v) canonicalizes around
// llvm.minnum/maxnum no matter what FP pragmas say; inline asm operands are
// opaque, so these lower to exactly one instruction each.
static __device__ __forceinline__ float vmin(float a, float b) {
    float r; asm("v_min_num_f32 %0, %1, %2" : "=v"(r) : "v"(a), "v"(b)); return r;
}
static __device__ __forceinline__ float vmin3(float a, float b, float c) {
    float r; asm("v_min3_num_f32 %0, %1, %2, %3" : "=v"(r) : "v"(a), "v"(b), "v"(c)); return r;
}
static __device__ __forceinline__ float vmax0(float a) {  // max(a, 0)
    float r; asm("v_max_num_f32 %0, 0, %1" : "=v"(r) : "v"(a)); return r;
}

// ---------------- init workspace (must run every call: graph replays) -------
__global__ void chamfer_init_ws(float* rowmin, float* sums, float* out) {
    int i = blockIdx.x * blockDim.x + threadIdx.x;
    if (i < BATCH * NBINS) rowmin[i] = BIGF;
    if (i < 2 * BATCH)     sums[i]   = 0.0f;
    if (i == 0)            out[0]    = 0.0f;
}

// ---------------- main: WMMA pairwise distance + dual min reduction ---------
// Quadratic-form trick, validity mask folded into the spare K slot:
//   A[n,:] = (x^2, x, 1, 1);  B[:,m] = (1, -2y, y^2, valid ? 0 : BIG)
//   D[n,m] = (x-y)^2 + (valid ? 0 : BIG)
// Tiles are processed in 2x2 blocks (t0,t1 x c0,c1) so every reduction step
// is a 3-input v_min3: 8 VALU ops per WMMA (2 tournaments, 2 elems per op).
__global__ void __launch_bounds__(256)
chamfer_main(const float* __restrict__ X,      // [B,256] bin centers
             const float* __restrict__ Y,      // [B,76800] target depths
             float*       __restrict__ rowmin, // [B*256] (ordered-uint min)
             float*       __restrict__ sums)   // [sumy[4] | cnty[4]]
{
    const int lane   = threadIdx.x & 31;
    const int waveId = blockIdx.x * (blockDim.x >> 5) + (threadIdx.x >> 5);
    const int b      = waveId / WPB;
    const int widx   = waveId % WPB;
    const int col16  = lane & 15;
    const bool lo    = lane < 16;

    // Preload A tiles for all 16 row-tiles of this batch's 256 bins.
    // Lanes 0-15 hold K=0,1 (x^2, x); lanes 16-31 hold K=2,3 (1, 1).
    float a0[16], a1[16];
#pragma unroll
    for (int t = 0; t < 16; ++t) {
        float x = X[b * NBINS + t * 16 + col16];
        a0[t] = lo ? x * x : 1.0f;
        a1[t] = lo ? x     : 1.0f;
    }

    // Running per-row minima (mask already applied inside D).
    float rm[128];
#pragma unroll
    for (int i = 0; i < 128; ++i) rm[i] = BIGF;

    float sumAcc = 0.0f, cntAcc = 0.0f;
    const float* Yb = Y + (size_t)b * MPTS;

    for (int c = 0; c < CTPW; c += 2) {
        const int m0 = (widx * CTPW + c) * 16;
        float y0 = Yb[m0 + col16];
        float y1 = Yb[m0 + 16 + col16];
        const bool valid0 = (y0 >= MIN_DEPTH);
        const bool valid1 = (y1 >= MIN_DEPTH);
        v2f bv0, bv1;
        bv0.x = lo ? 1.0f       : y0 * y0;
        bv0.y = lo ? -2.0f * y0 : (valid0 ? 0.0f : BIGF);
        bv1.x = lo ? 1.0f       : y1 * y1;
        bv1.y = lo ? -2.0f * y1 : (valid1 ? 0.0f : BIGF);

        float cmin0[8], cmin1[8];
#pragma unroll
        for (int tp = 0; tp < 8; ++tp) {
            const int t0 = 2 * tp, t1 = 2 * tp + 1;
            v2f av0; av0.x = a0[t0]; av0.y = a1[t0];
            v2f av1; av1.x = a0[t1]; av1.y = a1[t1];
            v8f z = {};
            v8f d00 = __builtin_amdgcn_wmma_f32_16x16x4_f32(
                false, av0, false, bv0, (short)0, z, false, false);
            v8f d01 = __builtin_amdgcn_wmma_f32_16x16x4_f32(
                false, av0, false, bv1, (short)0, z, false, false);
            v8f d10 = __builtin_amdgcn_wmma_f32_16x16x4_f32(
                false, av1, false, bv0, (short)0, z, false, false);
            v8f d11 = __builtin_amdgcn_wmma_f32_16x16x4_f32(
                false, av1, false, bv1, (short)0, z, false, false);
#pragma unroll
            for (int v = 0; v < 8; ++v) {
                // x->y: same row, both column tiles in one min3
                rm[t0 * 8 + v] = vmin3(rm[t0 * 8 + v], d00[v], d01[v]);
                rm[t1 * 8 + v] = vmin3(rm[t1 * 8 + v], d10[v], d11[v]);
                // y->x: same column tile, both row tiles in one min3
                if (tp == 0) {
                    cmin0[v] = vmin(d00[v], d10[v]);
                    cmin1[v] = vmin(d01[v], d11[v]);
                } else {
                    cmin0[v] = vmin3(cmin0[v], d00[v], d10[v]);
                    cmin1[v] = vmin3(cmin1[v], d01[v], d11[v]);
                }
            }
        }

        // Column min: 8 -> 1 via min3 tree, then across lane halves.
        float cm0 = vmin3(cmin0[0], cmin0[1], cmin0[2]);
        cm0 = vmin3(cm0, cmin0[3], cmin0[4]);
        cm0 = vmin3(cm0, cmin0[5], cmin0[6]);
        cm0 = vmin(cm0, cmin0[7]);
        float cm1 = vmin3(cmin1[0], cmin1[1], cmin1[2]);
        cm1 = vmin3(cm1, cmin1[3], cmin1[4]);
        cm1 = vmin3(cm1, cmin1[5], cmin1[6]);
        cm1 = vmin(cm1, cmin1[7]);
        cm0 = vmin(cm0, __shfl_xor(cm0, 16, 32));
        cm1 = vmin(cm1, __shfl_xor(cm1, 16, 32));
        if (lo && valid0) { sumAcc += cm0; cntAcc += 1.0f; }
        if (lo && valid1) { sumAcc += cm1; cntAcc += 1.0f; }
    }

    // y->x sums: full-wave reduce, one atomicAdd per wave.
#pragma unroll
    for (int off = 16; off >= 1; off >>= 1) {
        sumAcc += __shfl_xor(sumAcc, off, 32);
        cntAcc += __shfl_xor(cntAcc, off, 32);
    }
    if (lane == 0) {
        atomicAdd(&sums[b], sumAcc);
        atomicAdd(&sums[BATCH + b], cntAcc);
    }

    // x->y row minima: clamp once (makes ordered-uint atomicMin exact for the
    // non-negative domain), 16-lane min tree, then one atomic per row.
    unsigned int* rowminU = (unsigned int*)rowmin;
#pragma unroll
    for (int t = 0; t < 16; ++t) {
#pragma unroll
        for (int v = 0; v < 8; ++v) {
            float r = vmax0(rm[t * 8 + v]);
            r = vmin(r, __shfl_xor(r, 1, 32));
            r = vmin(r, __shfl_xor(r, 2, 32));
            r = vmin(r, __shfl_xor(r, 4, 32));
            r = vmin(r, __shfl_xor(r, 8, 32));
            if (lane == 0 || lane == 16) {
                int row = b * NBINS + t * 16 + v + (lo ? 0 : 8);
                atomicMin(&rowminU[row], __float_as_uint(r));
            }
        }
    }
}

// ---------------- finish: fold into scalar loss -----------------------------
__global__ void chamfer_final(const float* __restrict__ rowmin,
                              const float* __restrict__ sums,
                              float* __restrict__ out)
{
    __shared__ float s_cham[BATCH];
    const int lane = threadIdx.x & 31;
    const int w    = threadIdx.x >> 5;
    if (w < BATCH) {
        float s = 0.0f;
#pragma unroll
        for (int i = 0; i < NBINS / 32; ++i)
            s += rowmin[w * NBINS + i * 32 + lane];
#pragma unroll
        for (int off = 16; off >= 1; off >>= 1)
            s += __shfl_xor(s, off, 32);
        if (lane == 0) s_cham[w] = s / (float)NBINS;
    }
    __syncthreads();
    if (threadIdx.x == 0) {
        float loss = 0.0f;
        for (int bb = 0; bb < BATCH; ++bb)
            loss += s_cham[bb] + sums[bb] / sums[BATCH + bb];
        out[0] = 0.1f * loss * (1.0f / (float)BATCH);
    }
}

extern "C" void kernel_launch(void* const* d_in, const int* in_sizes, int n_in,
                              void* d_out, int out_size, void* d_ws, size_t ws_size,
                              hipStream_t stream) {
    (void)in_sizes; (void)n_in; (void)out_size; (void)ws_size;
    const float* bins = (const float*)d_in[0];   // [4,256,1] fp32
    const float* tgt  = (const float*)d_in[1];   // [4,240,320] fp32
    float* out    = (float*)d_out;               // scalar fp32
    float* rowmin = (float*)d_ws;                // 1024 floats
    float* sums   = rowmin + BATCH * NBINS;      // 8 floats

    chamfer_init_ws<<<(BATCH * NBINS + 255) / 256, 256, 0, stream>>>(rowmin, sums, out);

    // 4 batches x 150 waves = 600 waves; 8 waves (256 thr) per block -> 75 blocks
    chamfer_main<<<(BATCH * WPB * 32) / 256, 256, 0, stream>>>(bins, tgt, rowmin, sums);

    chamfer_final<<<1, 128, 0, stream>>>(rowmin, sums, out);
}